// Conv_62766652063747
// MI455X (gfx1250) — compile-verified
//
#include <hip/hip_runtime.h>

// f32 WMMA fragments (wave32): A 16x4 -> 2 VGPRs/lane, C/D 16x16 -> 8 VGPRs/lane
typedef float v2f __attribute__((ext_vector_type(2)));
typedef float v8f __attribute__((ext_vector_type(8)));

#define B_IMG   64
#define C_IN    3
#define H_IN    224
#define W_IN    224
#define HW_IN   (H_IN * W_IN)        // 50176
#define C_OUT   16
#define H_OUT   222
#define W_OUT   222
#define HW_OUT  (H_OUT * W_OUT)      // 49284
#define KTOT    27                   // C_IN*3*3, padded to 28 (7 chunks of K=4)
#define NCHUNK  7
#define TILES_PER_WAVE   6
#define WAVES_PER_BLOCK  8
#define THREADS (WAVES_PER_BLOCK * 32)

// Gather one input-patch element: k = ci*9 + kh*3 + kw, px already points at
// x[b][0][oh][ow].
__device__ __forceinline__ float patch_load(const float* __restrict__ px, int k) {
    int ci = k / 9;
    int r9 = k - ci * 9;
    int kh = r9 / 3;
    int kw = r9 - kh * 3;
    return px[ci * HW_IN + kh * W_IN + kw];
}

__global__ __launch_bounds__(THREADS)
void Conv_62766652063747_kernel(const float* __restrict__ x,
                                const float* __restrict__ w,   // [16][27] OIHW flat
                                float* __restrict__ out) {
    const int lane  = threadIdx.x & 31;
    const int nlo   = lane & 15;
    const int khalf = (lane >= 16) ? 2 : 0;       // K sub-offset for A/B fragments
    const int wu    = blockIdx.x * WAVES_PER_BLOCK + (threadIdx.x >> 5);

    // ---- A fragment: filter matrix, M = co = lane&15, K = 4c + khalf + {0,1} ----
    v2f aw[NCHUNK];
    {
        const float* wrow = w + nlo * KTOT;
        #pragma unroll
        for (int c = 0; c < NCHUNK; ++c) {
            int k0 = c * 4 + khalf;               // always <= 26
            aw[c].x = wrow[k0];
            int k1 = k0 + 1;
            float y = wrow[(k1 < KTOT) ? k1 : 0]; // clamped (non-divergent) load
            aw[c].y = (k1 < KTOT) ? y : 0.0f;     // zero-pad K=27
        }
    }

    const int pix_base = wu * (TILES_PER_WAVE * 16);

    for (int t = 0; t < TILES_PER_WAVE; ++t) {
        // N = pixel index within tile = lane&15 (matches B and C/D layouts)
        int pix  = pix_base + t * 16 + nlo;
        int bimg = pix / HW_OUT;
        int rem  = pix - bimg * HW_OUT;           // oh*222 + ow
        int oh   = rem / W_OUT;
        int ow   = rem - oh * W_OUT;
        const float* px = x + (bimg * C_IN) * HW_IN + oh * W_IN + ow;

        v8f acc = {};
        #pragma unroll
        for (int c = 0; c < NCHUNK; ++c) {
            int k0 = c * 4 + khalf;
            v2f bv;
            bv.x = patch_load(px, k0);
            int k1 = k0 + 1;
            float y = patch_load(px, (k1 < KTOT) ? k1 : 0);
            bv.y = (k1 < KTOT) ? y : 0.0f;
            // D = A(16x4 f32) * B(4x16 f32) + C  — full-precision f32 WMMA
            acc = __builtin_amdgcn_wmma_f32_16x16x4_f32(
                      /*neg_a=*/false, aw[c],
                      /*neg_b=*/false, bv,
                      /*c_mod=*/(short)0, acc,
                      /*reuse_a=*/false, /*reuse_b=*/false);
        }

        // C/D layout: VGPR r, lane l -> co = r + 8*(l>=16), N = l&15.
        // Per accumulator reg, each half-wave writes 16 consecutive floats (64B).
        int cobase = (lane >= 16) ? 8 : 0;
        float* orow = out + (bimg * C_OUT + cobase) * HW_OUT + rem;
        #pragma unroll
        for (int r = 0; r < 8; ++r) {
            orow[r * HW_OUT] = acc[r];
        }
    }
}

extern "C" void kernel_launch(void* const* d_in, const int* in_sizes, int n_in,
                              void* d_out, int out_size, void* d_ws, size_t ws_size,
                              hipStream_t stream) {
    const float* x = (const float*)d_in[0];   // [64,3,224,224] f32
    const float* w = (const float*)d_in[1];   // [16,3,3,3] f32
    float* out = (float*)d_out;               // [64,16,222,222] f32

    // 64*222*222 = 3,154,176 pixels = 197,136 tiles of 16
    //  = 32,856 wave-units of 6 tiles = 4,107 blocks of 8 waves. Exact, no tail.
    const int total_pixels = B_IMG * HW_OUT;
    const int tiles        = total_pixels / 16;
    const int wave_units   = tiles / TILES_PER_WAVE;
    const int blocks       = wave_units / WAVES_PER_BLOCK;

    Conv_62766652063747_kernel<<<blocks, THREADS, 0, stream>>>(x, w, out);
}